// Predicting_Base_45621142618503
// MI455X (gfx1250) — compile-verified
//
#include <hip/hip_runtime.h>
#include <hip/hip_bf16.h>
#include <stdint.h>

#define BB    16
#define NN    100000
#define DD    25
#define CC    20
#define KK    500
#define KP    512
#define THRC  0.05f
#define THRN  0.5f

typedef unsigned int v4u __attribute__((ext_vector_type(4)));
typedef int          v8i __attribute__((ext_vector_type(8)));
typedef int          v4i __attribute__((ext_vector_type(4)));

__device__ __forceinline__ float sigm(float x) { return 1.0f / (1.0f + __expf(-x)); }

// ---------------------------------------------------------------------------
// Kernel 1: double-buffered TDM pipeline. Each block owns a strip of 10
// tiles (256 detections x 100 B each); tile t+1 streams into the alternate
// LDS buffer while tile t is scored. s_wait_tensorcnt(1) exploits in-order
// TENSORcnt completion to synchronize only on the older transfer.
// ---------------------------------------------------------------------------
#define T1 256
#define TILE_DETS   T1
#define TILE_FLOATS (TILE_DETS * DD)   // 6400 floats = 25600 bytes per tile
#define TILES_PB    10                 // 6250 tiles total -> 625 blocks

__device__ __forceinline__ void tdm_issue(const float* in, uint64_t tileIdx,
                                          float* ldsBuf)
{
    uint64_t gaddr = (uint64_t)(uintptr_t)in + tileIdx * (uint64_t)(TILE_FLOATS * 4);
    // flat addr of LDS object: low 32 bits == LDS byte offset (ISA 10.2)
    uint32_t lds_off = (uint32_t)(uintptr_t)ldsBuf;

    v4u g0;
    g0[0] = 1u;                                          // count=1, user mode
    g0[1] = lds_off;                                     // lds_addr
    g0[2] = (uint32_t)(gaddr & 0xFFFFFFFFu);             // global_addr[31:0]
    g0[3] = (uint32_t)((gaddr >> 32) & 0x01FFFFFFu)      // global_addr[56:32]
          | (2u << 30);                                  // type=2 ("image")
    v8i g1;
    g1[0] = (int)(2u << 16);                             // data_size=4B
    g1[1] = (int)(((unsigned)TILE_FLOATS & 0xFFFFu) << 16); // tensor_dim0 lo16
    g1[2] = (int)(((unsigned)TILE_FLOATS >> 16) | (1u << 16)); // dim0 hi16 | tensor_dim1=1
    g1[3] = (int)((unsigned)TILE_FLOATS << 16);          // tile_dim0 = 6400
    g1[4] = 0;                                           // tile_dim1=0 (1-D), tile_dim2=0
    g1[5] = TILE_FLOATS;                                 // tensor_dim0_stride lo32
    g1[6] = 0;
    g1[7] = 0;
    v4i g2 = {0, 0, 0, 0};
    v4i g3 = {0, 0, 0, 0};
    v8i g4 = {0, 0, 0, 0, 0, 0, 0, 0};
    __builtin_amdgcn_tensor_load_to_lds(g0, g1, g2, g3, g4, 0);
}

__global__ void score_kernel(const float* __restrict__ in, uint32_t* __restrict__ keys)
{
    __shared__ float tile[2][TILE_FLOATS];               // 51.2 KB double buffer
    const unsigned tid = threadIdx.x;
    const uint64_t baseTile = (uint64_t)blockIdx.x * TILES_PB;

    if (tid < 32) tdm_issue(in, baseTile, &tile[0][0]);  // prologue: tile 0

    for (int t = 0; t < TILES_PB; ++t) {
        const int buf = t & 1;
        if (tid < 32) {
            if (t + 1 < TILES_PB) {
                tdm_issue(in, baseTile + t + 1, &tile[buf ^ 1][0]);
                __builtin_amdgcn_s_wait_tensorcnt(1);    // tile t landed (in-order)
            } else {
                __builtin_amdgcn_s_wait_tensorcnt(0);
            }
        }
        __syncthreads();

        // stride 25 (odd) across 64 LDS banks -> conflict-free
        const float* p = &tile[buf][tid * DD];
        float conf = sigm(p[4]);
        float m = p[5];
#pragma unroll
        for (int c = 1; c < CC; ++c) m = fmaxf(m, p[5 + c]);
        float s = 0.0f;
#pragma unroll
        for (int c = 0; c < CC; ++c) s += __expf(p[5 + c] - m);
        float score  = conf / s;             // sigmoid(conf) * max(softmax(cls))
        float masked = (score > THRC) ? score : 0.0f;
        keys[(baseTile + t) * TILE_DETS + tid] = __float_as_uint(masked);

        __syncthreads();                     // buffer reusable next iteration
    }
}

// ---------------------------------------------------------------------------
// Kernel 2: per-batch radix-select top-500 + bitonic sort + decode + NMS.
// One 512-thread workgroup per batch image.
// ---------------------------------------------------------------------------
#define T2 512

__global__ void select_nms_kernel(const float* __restrict__ in,
                                  const uint32_t* __restrict__ keys,
                                  float* __restrict__ out)
{
    const int b   = blockIdx.x;
    const int tid = threadIdx.x;
    const uint32_t* kb = keys + (size_t)b * NN;

    __shared__ uint32_t hist[256];
    __shared__ int      s_bin, s_cntHigh, s_cntEq;
    __shared__ uint32_t s_acc;
    __shared__ uint32_t candKey[KP];
    __shared__ int      candId[KP];

    // ---- 4-pass radix select: exact value of the 500th-largest key ----
    uint32_t prefix = 0, pmask = 0;
    int remaining = KK;
    for (int shift = 24; shift >= 0; shift -= 8) {
        for (int i = tid; i < 256; i += T2) hist[i] = 0;
        __syncthreads();
        for (int i = tid; i < NN; i += T2) {
            uint32_t k = kb[i];
            if ((k & pmask) == prefix) atomicAdd(&hist[(k >> shift) & 255u], 1u);
        }
        __syncthreads();
        if (tid == 0) {
            uint32_t acc = 0; int bin = 0;
            for (int d = 255; d >= 0; --d) {
                uint32_t c = hist[d];
                if (acc + c >= (uint32_t)remaining) { bin = d; break; }
                acc += c;
            }
            s_bin = bin; s_acc = acc;
        }
        __syncthreads();
        prefix |= ((uint32_t)s_bin) << shift;
        pmask  |= 0xFFu << shift;
        remaining -= (int)s_acc;
        __syncthreads();
    }
    const uint32_t thr = prefix;

    // ---- compact: all keys > thr (count < 500), then fill with == thr ----
    if (tid == 0) { s_cntHigh = 0; s_cntEq = 0; }
    for (int i = tid; i < KP; i += T2) { candKey[i] = 0u; candId[i] = 0; }
    __syncthreads();
    for (int i = tid; i < NN; i += T2) {
        uint32_t k = kb[i];
        if (k > thr) {
            int p = atomicAdd(&s_cntHigh, 1);
            if (p < KK) { candKey[p] = k; candId[p] = i; }
        }
    }
    __syncthreads();
    int nHigh = (s_cntHigh < KK) ? s_cntHigh : KK;
    for (int i = tid; i < NN; i += T2) {
        uint32_t k = kb[i];
        if (k == thr) {
            int p = atomicAdd(&s_cntEq, 1);
            int slot = nHigh + p;
            if (slot < KK) { candKey[slot] = k; candId[slot] = i; }
        }
    }
    __syncthreads();

    // ---- bitonic sort 512 entries: key desc, id asc ----
    for (unsigned seg = 2; seg <= KP; seg <<= 1) {
        for (unsigned j = seg >> 1; j > 0; j >>= 1) {
            unsigned i = (unsigned)tid, ixj = i ^ j;
            if (ixj > i) {
                uint32_t ka = candKey[i], kc = candKey[ixj];
                int ia = candId[i], ic = candId[ixj];
                bool before = (ka > kc) || (ka == kc && ia <= ic);
                bool up = ((i & seg) == 0);
                if (up ? !before : before) {
                    candKey[i] = kc; candKey[ixj] = ka;
                    candId[i]  = ic; candId[ixj]  = ia;
                }
            }
            __syncthreads();
        }
    }

    // ---- gather winners: decode boxes, argmax label ----
    __shared__ float bl[KK], bt[KK], br[KK], bo[KK], ar[KK], sc[KK];
    __shared__ int   lab[KK];
    __shared__ unsigned char validArr[KK], keepArr[KK];
    if (tid < KK) {
        int id = candId[tid];
        const float* p = in + ((size_t)b * NN + id) * DD;
        float cx = sigm(p[0]), cy = sigm(p[1]);
        float w  = sigm(p[2]), h  = sigm(p[3]);
        float l = cx - 0.5f * w, t = cy - 0.5f * h;
        float r = cx + 0.5f * w, d = cy + 0.5f * h;
        int lbl = 0; float mv = p[5];
        for (int c = 1; c < CC; ++c) { float v = p[5 + c]; if (v > mv) { mv = v; lbl = c; } }
        float score = __uint_as_float(candKey[tid]);
        bl[tid] = l; bt[tid] = t; br[tid] = r; bo[tid] = d;
        ar[tid] = fmaxf(r - l, 0.0f) * fmaxf(d - t, 0.0f);
        lab[tid] = lbl; sc[tid] = score;
        validArr[tid] = (score > THRC) ? 1 : 0;
        float* ob = out + (size_t)BB * KK + ((size_t)b * KK + tid) * 4;  // boxes1
        ob[0] = l; ob[1] = t; ob[2] = r; ob[3] = d;
    }
    __syncthreads();

    // ---- suppression bit-matrix: sup[i] has bit j set if (j<i) suppresses i ----
    __shared__ uint32_t sup[KK][16];
    for (int i = tid; i < KK; i += T2) {
        float li = bl[i], ti = bt[i], ri = br[i], di = bo[i], ai = ar[i];
        int   Li = lab[i];
        for (int w = 0; w < 16; ++w) {
            uint32_t m = 0;
            int jbase = w * 32;
            for (int jo = 0; jo < 32; ++jo) {
                int j = jbase + jo;
                if (j >= i || j >= KK) break;
                if (lab[j] == Li) {
                    float xl = fmaxf(li, bl[j]), yt = fmaxf(ti, bt[j]);
                    float xr = fminf(ri, br[j]), yb = fminf(di, bo[j]);
                    float iw = fmaxf(xr - xl, 0.0f), ih = fmaxf(yb - yt, 0.0f);
                    float inter = iw * ih;
                    float uni   = ai + ar[j] - inter;
                    if (inter / fmaxf(uni, 1e-9f) > THRN) m |= (1u << jo);
                }
            }
            sup[i][w] = m;
        }
    }
    __syncthreads();

    // ---- greedy scan over kept-bitmask (serial but only ~8K bit-ops) ----
    if (tid == 0) {
        uint32_t kept[16] = {0};
        for (int i = 0; i < KK; ++i) {
            uint32_t any = 0;
            for (int w = 0; w < 16; ++w) any |= sup[i][w] & kept[w];
            bool ki = validArr[i] && (any == 0u);
            keepArr[i] = ki ? 1 : 0;
            if (ki) kept[i >> 5] |= 1u << (i & 31);
        }
    }
    __syncthreads();

    // ---- emit ids_batch2 / labels2 / scores2 ----
    if (tid < KK) {
        bool ki = keepArr[tid] != 0;
        size_t idx = (size_t)b * KK + tid;
        out[idx]                        = (float)b;                       // ids_batch2
        out[5 * (size_t)BB * KK + idx]  = ki ? (float)lab[tid] : -1.0f;   // labels2
        out[6 * (size_t)BB * KK + idx]  = ki ? sc[tid] : 0.0f;            // scores2
    }
}

// ---------------------------------------------------------------------------
extern "C" void kernel_launch(void* const* d_in, const int* in_sizes, int n_in,
                              void* d_out, int out_size, void* d_ws, size_t ws_size,
                              hipStream_t stream)
{
    const float* outs = (const float*)d_in[0];
    uint32_t* keys = (uint32_t*)d_ws;      // B*N u32 keys = 6.4 MB scratch
    float* out = (float*)d_out;

    // 16*100000*25 floats == 6250 tiles; 10 tiles per block -> 625 blocks
    score_kernel<<<(BB * NN) / (T1 * TILES_PB), T1, 0, stream>>>(outs, keys);
    select_nms_kernel<<<BB, T2, 0, stream>>>(outs, keys, out);
}